// KPlaneField_71399536329012
// MI455X (gfx1250) — compile-verified
//
#include <hip/hip_runtime.h>
#include <stdint.h>

typedef float f4 __attribute__((ext_vector_type(4)));

#define FDIM 16
#define WRES 64

__device__ __forceinline__ int iclamp(int v, int lo, int hi) {
  return v < lo ? lo : (v > hi ? hi : v);
}

// ---------------------------------------------------------------------------
// Pass 1: layout transform  G[p][f][H][64]  ->  T[p][H][64][f]
//
// TDM version: each tile is (plane p, row y): 16 feature-rows of 64 floats,
// row stride H*64 elements -> one tensor_load_to_lds 2D descriptor lands the
// 4KB tile contiguously in LDS. Double-buffered per wave with
// s_wait_tensorcnt. LDS is then read transposed via ds_load_b32 (explicit
// addrspace(3) volatile pointer) and written out as coalesced b128 stores
// (the 4KB output block is contiguous).
// ---------------------------------------------------------------------------

#define NTILES 5760   // sum over grids of 3*H = 3*(128+256+512+1024)

__global__ void kplane_transpose_tdm(const float* __restrict__ G0,
                                     const float* __restrict__ G1,
                                     const float* __restrict__ G2,
                                     const float* __restrict__ G3,
                                     float* __restrict__ T,
                                     int tstride) {
#if defined(__HIP_DEVICE_COMPILE__) && \
    __has_builtin(__builtin_amdgcn_tensor_load_to_lds) && \
    __has_builtin(__builtin_amdgcn_s_wait_tensorcnt)
  typedef unsigned int u32x4 __attribute__((ext_vector_type(4)));
  typedef int i32x8 __attribute__((ext_vector_type(8)));
  typedef int i32x4 __attribute__((ext_vector_type(4)));
  typedef __attribute__((address_space(3))) volatile const float* ldsv_t;

  __shared__ float lds[16384];                 // 8 waves * 2 buffers * 1024 f
  const int lane = threadIdx.x & 31;
  const int wslot = threadIdx.x >> 5;          // wave in block (0..7)
  int t = blockIdx.x * (blockDim.x >> 5) + wslot;
  const unsigned ldsOff0 =
      (unsigned)(uintptr_t)(&lds[0]) + (unsigned)(wslot * 8192);

  const float* Gs[4] = {G0, G1, G2, G3};

  // --- issue helper (lambda): one 2D TDM tile load ---
  auto issue = [&](int tt, int buf) {
    int g = (tt >= 2688) ? 3 : (tt >= 1152) ? 2 : (tt >= 384) ? 1 : 0;
    int m = (1 << g) - 1;
    int local = tt - 384 * m;                   // cumb[g] = 384*(2^g-1)
    int H = 128 << g;
    int p = local >> (7 + g);
    int y = local & (H - 1);
    const float* src = Gs[g] + (size_t)p * 1024u * (size_t)H + (size_t)y * 64u;
    unsigned long long ga = (unsigned long long)(uintptr_t)src;

    u32x4 d0;
    d0.x = 1u;                                  // count=1, user descriptor
    d0.y = ldsOff0 + (unsigned)(buf * 4096);    // LDS byte address
    d0.z = (unsigned)(ga & 0xffffffffu);        // global_addr[31:0]
    d0.w = (unsigned)((ga >> 32) & 0x01ffffffu) // global_addr[56:32]
           | (2u << 30);                        // type = 2 ("image")
    i32x8 d1;
    d1[0] = (2 << 16);                          // data_size = 4B
    d1[1] = (64 << 16);                         // tensor_dim0 = 64
    d1[2] = (16 << 16);                         // tensor_dim1 = 16
    d1[3] = (64 << 16);                         // tile_dim0 = 64
    d1[4] = 16;                                 // tile_dim1 = 16, tile_dim2=0
    d1[5] = H * 64;                             // tensor_dim0_stride (elems)
    d1[6] = 0;
    d1[7] = 0;
    i32x4 z4 = {0, 0, 0, 0};
#if __clang_major__ >= 23
    i32x8 z8 = {0, 0, 0, 0, 0, 0, 0, 0};
    __builtin_amdgcn_tensor_load_to_lds(d0, d1, z4, z4, z8, 0);
#else
    __builtin_amdgcn_tensor_load_to_lds(d0, d1, z4, z4, 0);
#endif
  };

  int buf = 0;
  if (t < NTILES) issue(t, buf);
  while (t < NTILES) {
    int tn = t + tstride;
    if (tn < NTILES) {
      issue(tn, buf ^ 1);
      __builtin_amdgcn_s_wait_tensorcnt(1);     // oldest (current buf) done
    } else {
      __builtin_amdgcn_s_wait_tensorcnt(0);
    }
    asm volatile("" ::: "memory");              // keep ds reads after wait

    // ---- drain tile t from LDS, transposed, as coalesced b128 stores ----
    int g = (t >= 2688) ? 3 : (t >= 1152) ? 2 : (t >= 384) ? 1 : 0;
    int m = (1 << g) - 1;
    int local = t - 384 * m;
    int H = 128 << g;
    int p = local >> (7 + g);
    int y = local & (H - 1);
    float* outp = T + (size_t)393216u * (size_t)m   // goff[g] = 393216*(2^g-1)
                    + ((size_t)((p << (7 + g)) + y)) * 1024u;
    ldsv_t lb = (ldsv_t)(ldsOff0 + (unsigned)(buf * 4096));

    #pragma unroll
    for (int c8 = 0; c8 < 8; ++c8) {
      int chunk = (c8 << 5) + lane;             // 0..255 float4 chunks
      int x = chunk >> 2;
      int f0 = (chunk & 3) << 2;
      f4 v;
      v.x = lb[(f0 + 0) * 64 + x];              // ds_load_b32
      v.y = lb[(f0 + 1) * 64 + x];
      v.z = lb[(f0 + 2) * 64 + x];
      v.w = lb[(f0 + 3) * 64 + x];
      *(f4*)(outp + ((size_t)chunk << 2)) = v;  // RT store: keep table in L2
    }
    buf ^= 1;
    t = tn;
  }
#else
  // Fallback body (toolchains without the TDM builtin): elementwise.
  const float* Gs[4] = {G0, G1, G2, G3};
  int tid = blockIdx.x * blockDim.x + threadIdx.x;
  int nthreads = blockDim.x * gridDim.x;
  (void)tstride;
  for (int i = tid; i < 5898240; i += nthreads) {
    int g = (i >= 2752512) ? 3 : (i >= 1179648) ? 2 : (i >= 393216) ? 1 : 0;
    int m = (1 << g) - 1;
    int local = i - 393216 * m;
    int f = local & 15;
    int x = (local >> 4) & 63;
    int yp = local >> 10;
    int H = 128 << g;
    int y = yp & (H - 1);
    int p = yp >> (7 + g);
    T[i] = Gs[g][(((size_t)(p * FDIM + f) * (size_t)H) + (size_t)y) * WRES + x];
  }
#endif
}

// ---------------------------------------------------------------------------
// Pass 2: gather + bilinear blend.
// Block = 192 threads = 4 points x 48 float4-slots.
// slot j -> (g = j/12, p = (j%12)/4, f0 = (j%4)*4); output offset = 16B * j,
// so each wave32's store instruction covers 512B contiguous. Gather loads are
// b128 with 4 consecutive lanes covering a contiguous 64B feature run.
// Output stream (768MB) uses NT stores to avoid evicting the gather table.
// ---------------------------------------------------------------------------
__global__ void kplane_gather(const float* __restrict__ pts,
                              const float* __restrict__ radius,
                              const float* __restrict__ T,
                              float* __restrict__ out, int N) {
  int lane = threadIdx.x;
  int n = blockIdx.x * 4 + (lane / 48);
  int j = lane % 48;
  if (n >= N) return;

  int g   = j / 12;
  int rem = j - g * 12;
  int p   = rem >> 2;
  int f0  = (rem & 3) << 2;

  float coord = pts[(size_t)n * 3 + p];
  float rad   = radius[n];

  int H = 128 << g;
  float xf  = (coord + 1.0f) * 0.5f * 63.0f;
  float yf  = (rad + 1.0f) * 0.5f * (float)(H - 1);
  float x0f = floorf(xf);
  float y0f = floorf(yf);
  float wx  = xf - x0f;
  float wy  = yf - y0f;
  int ix = (int)x0f, iy = (int)y0f;
  int x0 = iclamp(ix, 0, 63);
  int x1 = iclamp(ix + 1, 0, 63);
  int y0 = iclamp(iy, 0, H - 1);
  int y1 = iclamp(iy + 1, 0, H - 1);

  const float* P = T + (size_t)393216u * (size_t)((1u << g) - 1u)
                     + (size_t)p * ((size_t)H << 10);
  const float* a00 = P + ((((size_t)y0 << 6) + (size_t)x0) << 4) + f0;
  const float* a01 = P + ((((size_t)y0 << 6) + (size_t)x1) << 4) + f0;
  const float* a10 = P + ((((size_t)y1 << 6) + (size_t)x0) << 4) + f0;
  const float* a11 = P + ((((size_t)y1 << 6) + (size_t)x1) << 4) + f0;

  __builtin_prefetch(a10, 0, 1);   // gfx1250: global_prefetch_b8
  __builtin_prefetch(a11, 0, 1);

  f4 v00 = *(const f4*)a00;
  f4 v01 = *(const f4*)a01;
  f4 v10 = *(const f4*)a10;
  f4 v11 = *(const f4*)a11;

  float w00 = (1.0f - wy) * (1.0f - wx);
  float w01 = (1.0f - wy) * wx;
  float w10 = wy * (1.0f - wx);
  float w11 = wy * wx;

  f4 o = v00 * w00 + v01 * w01 + v10 * w10 + v11 * w11;

  __builtin_nontemporal_store(o, (f4*)(out + (size_t)n * 192 + (size_t)(j << 2)));
}

// ---------------------------------------------------------------------------
// Fallback (ws too small): gather straight from original [p][f][H][W] layout.
// ---------------------------------------------------------------------------
__global__ void kplane_gather_direct(const float* __restrict__ pts,
                                     const float* __restrict__ radius,
                                     const float* __restrict__ G0,
                                     const float* __restrict__ G1,
                                     const float* __restrict__ G2,
                                     const float* __restrict__ G3,
                                     float* __restrict__ out, int N) {
  int n = blockIdx.x;
  int j = threadIdx.x;
  if (n >= N) return;
  int g   = j / 48;
  int rem = j % 48;
  int p   = rem >> 4;
  int f   = rem & 15;
  const float* G = (g == 0) ? G0 : (g == 1) ? G1 : (g == 2) ? G2 : G3;
  int H = 128 << g;

  float coord = pts[(size_t)n * 3 + p];
  float rad   = radius[n];
  float xf  = (coord + 1.0f) * 0.5f * 63.0f;
  float yf  = (rad + 1.0f) * 0.5f * (float)(H - 1);
  float x0f = floorf(xf);
  float y0f = floorf(yf);
  float wx  = xf - x0f;
  float wy  = yf - y0f;
  int ix = (int)x0f, iy = (int)y0f;
  int x0 = iclamp(ix, 0, 63);
  int x1 = iclamp(ix + 1, 0, 63);
  int y0 = iclamp(iy, 0, H - 1);
  int y1 = iclamp(iy + 1, 0, H - 1);

  size_t base = (size_t)(p * FDIM + f) * (size_t)H;
  float v00 = G[(base + y0) * WRES + x0];
  float v01 = G[(base + y0) * WRES + x1];
  float v10 = G[(base + y1) * WRES + x0];
  float v11 = G[(base + y1) * WRES + x1];

  float o = v00 * (1.0f - wy) * (1.0f - wx) + v01 * (1.0f - wy) * wx +
            v10 * wy * (1.0f - wx) + v11 * wy * wx;
  __builtin_nontemporal_store(o, out + (size_t)n * 192 + j);
}

extern "C" void kernel_launch(void* const* d_in, const int* in_sizes, int n_in,
                              void* d_out, int out_size, void* d_ws, size_t ws_size,
                              hipStream_t stream) {
  const float* pts    = (const float*)d_in[0];
  const float* radius = (const float*)d_in[1];
  const float* G[4]   = {(const float*)d_in[2], (const float*)d_in[3],
                         (const float*)d_in[4], (const float*)d_in[5]};
  float* out = (float*)d_out;
  int N = in_sizes[0] / 3;

  const size_t needFloats = 5898240;            // 3*16*64*128*(1+2+4+8)
  const size_t needBytes  = needFloats * 4;     // 23,592,960 B

  if (ws_size >= needBytes) {
    float* T = (float*)d_ws;
    const int tblocks = 360;                    // 2880 waves, 2 tiles each
    kplane_transpose_tdm<<<tblocks, 256, 0, stream>>>(G[0], G[1], G[2], G[3],
                                                      T, tblocks * 8);
    int blocks = (N + 3) / 4;
    kplane_gather<<<blocks, 192, 0, stream>>>(pts, radius, T, out, N);
  } else {
    kplane_gather_direct<<<N, 192, 0, stream>>>(pts, radius, G[0], G[1], G[2], G[3],
                                                out, N);
  }
}